// Generater_80762565034318
// MI455X (gfx1250) — compile-verified
//
#include <hip/hip_runtime.h>

// ---------------------------------------------------------------------------
// CDNA5 (gfx1250) implementation of the "Generater" network.
// All GEMM-like work (convs via implicit GEMM, attention score & AV GEMMs)
// runs through V_WMMA_F32_16X16X4_F32 (wave32). Each wave owns a 16x64 C tile
// (four 16x16 WMMA accumulators sharing one A fragment). Attention GEMMs are
// arranged so both operands have unit K-stride -> paired b64 vector loads.
// Softmax normalization is fused into the AV GEMM store (per-column scale).
// ---------------------------------------------------------------------------

typedef __attribute__((ext_vector_type(2))) float v2f;
typedef __attribute__((ext_vector_type(8))) float v8f;

__device__ __forceinline__ v8f wmma4(v8f acc, float a0, float a1, float b0, float b1) {
  v2f a; a[0] = a0; a[1] = a1;
  v2f b; b[0] = b0; b[1] = b1;
  return __builtin_amdgcn_wmma_f32_16x16x4_f32(
      false, a, false, b, (short)0, acc, false, false);
}

__device__ __forceinline__ v8f wmma4v(v8f acc, v2f a, v2f b) {
  return __builtin_amdgcn_wmma_f32_16x16x4_f32(
      false, a, false, b, (short)0, acc, false, false);
}

// ---------------------------------------------------------------------------
// Implicit-GEMM convolution. One wave computes a 16(M=Cout) x 64(N=B*OH*OW)
// tile. Reduction nested (kh,kw) outer, ci inner: hot-loop addressing is pure
// pointer increments. A rows clamped for m>=Cout, B columns clamped for n>=N
// (those C entries are never stored); only the K(=ci) tail is zero-gated
// (matters only for the Cin=3 stem conv).
// ---------------------------------------------------------------------------
__global__ void __launch_bounds__(128)
conv_wmma_kernel(const float* __restrict__ in, const float* __restrict__ wgt,
                 const float* __restrict__ resid, float* __restrict__ out,
                 int B, int Cin, int Hin, int Win, int Cout, int Hout, int Wout,
                 int KH, int KW, int stride, int pad, int relu) {
  const int lane = threadIdx.x & 31;
  const int wave = threadIdx.x >> 5;
  const int half = lane >> 4;   // 0: K=k0,k0+1  1: K=k0+2,k0+3
  const int l16  = lane & 15;

  const int N    = B * Hout * Wout;
  const int K    = Cin * KH * KW;
  const int HW   = Hin * Win;
  const int KHKW = KH * KW;
  const int nT   = (N + 63) >> 6;
  const int mT   = (Cout + 15) >> 4;
  const int tile = blockIdx.x * 4 + wave;           // 4 waves / block
  if (tile >= mT * nT) return;                      // uniform per wave
  const int mt = tile / nT, nt = tile % nT;
  const int mBase = mt << 4, nBase = nt << 6;

  const int m = mBase + l16;
  const float* aRow = wgt + (size_t)min(m, Cout - 1) * K;

  int ow[4], oh[4], bb[4], ih0[4], iw0[4];
  bool nok[4];
#pragma unroll
  for (int s = 0; s < 4; ++s) {
    const int n = nBase + 16 * s + l16;
    nok[s] = (n < N);
    const int nc = nok[s] ? n : (N - 1);
    ow[s] = nc % Wout; int t = nc / Wout; oh[s] = t % Hout; bb[s] = t / Hout;
    ih0[s] = oh[s] * stride - pad;
    iw0[s] = ow[s] * stride - pad;
  }

  v8f z = {};
  v8f acc[4] = {z, z, z, z};
  for (int kh = 0; kh < KH; ++kh) {
    for (int kw = 0; kw < KW; ++kw) {
      const float* ap = aRow + kh * KW + kw;                 // + ci*KHKW
      const float* bp[4];
      bool ok[4];
#pragma unroll
      for (int s = 0; s < 4; ++s) {
        const int ih = ih0[s] + kh, iw = iw0[s] + kw;
        ok[s] = (unsigned)ih < (unsigned)Hin && (unsigned)iw < (unsigned)Win;
        bp[s] = in + ((size_t)bb[s] * Cin * Hin + ih) * Win + iw;   // + ci*HW
      }
#pragma unroll 2
      for (int ci = 0; ci < Cin; ci += 4) {
        const int c0 = ci + 2 * half;
        const int c1 = c0 + 1;
        const int c0c = min(c0, Cin - 1), c1c = min(c1, Cin - 1);
        const float a0 = ap[(size_t)c0c * KHKW];
        const float a1 = ap[(size_t)c1c * KHKW];
#pragma unroll
        for (int s = 0; s < 4; ++s) {
          const float b0 = (ok[s] && c0 < Cin) ? bp[s][(size_t)c0 * HW] : 0.f;
          const float b1 = (ok[s] && c1 < Cin) ? bp[s][(size_t)c1 * HW] : 0.f;
          acc[s] = wmma4(acc[s], a0, a1, b0, b1);
        }
      }
    }
  }

  // C/D layout: VGPR j -> M = mBase + j + 8*half, N column = lane's l16.
  for (int j = 0; j < 8; ++j) {
    const int mm = mBase + j + 8 * half;
    if (mm < Cout) {
#pragma unroll
      for (int s = 0; s < 4; ++s) {
        if (nok[s]) {
          const size_t idx = (((size_t)bb[s] * Cout + mm) * Hout + oh[s]) * Wout + ow[s];
          float v = acc[s][j];
          if (resid) v += resid[idx];
          if (relu) v = fmaxf(v, 0.f);
          out[idx] = v;
        }
      }
    }
  }
}

// ---------------------------------------------------------------------------
// Unit-K-stride GEMM: C[m,n] = sum_k A[m*lam + k] * B[n*lbn + k], with an
// optional per-column scale applied at the store (fused softmax denominator).
// Both operands contiguous along K -> each lane's (k0+2h, k0+2h+1) pair is
// one aligned b64 load. REQUIRES K % 4 == 0 and lam/lbn multiples of 2.
// ---------------------------------------------------------------------------
__global__ void __launch_bounds__(128)
gemm_kk_wmma_kernel(const float* __restrict__ A, const float* __restrict__ Bm,
                    float* __restrict__ C, int M, int N, int K,
                    int lam, int lbn, int lcm, int lcn,
                    const float* __restrict__ colScale, int relu) {
  const int lane = threadIdx.x & 31;
  const int wave = threadIdx.x >> 5;
  const int half = lane >> 4;
  const int l16  = lane & 15;

  const int nT = (N + 63) >> 6;
  const int mT = (M + 15) >> 4;
  const int tile = blockIdx.x * 4 + wave;
  if (tile >= mT * nT) return;
  const int mt = tile / nT, nt = tile % nT;
  const int mBase = mt << 4, nBase = nt << 6;
  const int m = mBase + l16;

  const float* ap = A + (size_t)min(m, M - 1) * lam + 2 * half;
  const float* bp[4];
  int nIdx[4];
#pragma unroll
  for (int s = 0; s < 4; ++s) {
    nIdx[s] = nBase + 16 * s + l16;
    bp[s] = Bm + (size_t)min(nIdx[s], N - 1) * lbn + 2 * half;
  }
  __builtin_prefetch(ap, 0, 1);
  __builtin_prefetch(bp[0], 0, 1);

  v8f z = {};
  v8f acc[4] = {z, z, z, z};
#pragma unroll 2
  for (int k0 = 0; k0 < K; k0 += 4) {
    const v2f a = *(const v2f*)ap; ap += 4;
#pragma unroll
    for (int s = 0; s < 4; ++s) {
      const v2f b = *(const v2f*)bp[s]; bp[s] += 4;
      acc[s] = wmma4v(acc[s], a, b);
    }
  }

  float csc[4];
#pragma unroll
  for (int s = 0; s < 4; ++s)
    csc[s] = colScale ? colScale[min(nIdx[s], N - 1)] : 1.f;

  for (int j = 0; j < 8; ++j) {
    const int mm = mBase + j + 8 * half;
    if (mm < M) {
#pragma unroll
      for (int s = 0; s < 4; ++s) {
        if (nIdx[s] < N) {
          float v = acc[s][j] * csc[s];
          if (relu) v = fmaxf(v, 0.f);
          C[(size_t)mm * lcm + (size_t)nIdx[s] * lcn] = v;
        }
      }
    }
  }
}

// ---------------------------------------------------------------------------
// Bilinear 2x upsample, align_corners=True, writing TRANSPOSED output:
// src [B][C][H][W] -> dst [B][4HW][C]  (query-major, channel-minor)
// ---------------------------------------------------------------------------
__global__ void bilinear_up2_t_kernel(const float* __restrict__ src, float* __restrict__ dst,
                                      int B, int C, int H, int W) {
  const int H2 = 2 * H, W2 = 2 * W, Mq = H2 * W2;
  const int total = B * C * Mq;
  int i = blockIdx.x * blockDim.x + threadIdx.x;
  if (i >= total) return;
  int x2 = i % W2; int t = i / W2; int y2 = t % H2; t /= H2; int c = t % C; int b = t / C;
  float yc = (H > 1) ? (float)y2 * (float)(H - 1) / (float)(H2 - 1) : 0.f;
  float xc = (W > 1) ? (float)x2 * (float)(W - 1) / (float)(W2 - 1) : 0.f;
  int y0 = (int)floorf(yc); int y1 = min(y0 + 1, H - 1);
  int x0 = (int)floorf(xc); int x1 = min(x0 + 1, W - 1);
  float wy = yc - (float)y0, wx = xc - (float)x0;
  const float* s = src + ((size_t)b * C + c) * H * W;
  float v = (1.f - wy) * ((1.f - wx) * s[y0 * W + x0] + wx * s[y0 * W + x1]) +
            wy         * ((1.f - wx) * s[y1 * W + x0] + wx * s[y1 * W + x1]);
  const int mq = y2 * W2 + x2;
  dst[((size_t)b * Mq + mq) * C + c] = v;
}

// src [B][C][P] -> dst [B][P][C]
__global__ void transpose_cp_kernel(const float* __restrict__ src, float* __restrict__ dst,
                                    int B, int C, int P) {
  const int total = B * C * P;
  int i = blockIdx.x * blockDim.x + threadIdx.x;
  if (i >= total) return;
  int p = i % P; int t = i / P; int c = t % C; int b = t / C;
  dst[((size_t)b * P + p) * C + c] = src[i];
}

// ---------------------------------------------------------------------------
// Softmax exp pass: one block per row of length P (<=1024). Each thread holds
// its <=4 row elements in registers (single global read + single write).
// Writes un-normalized exp(s - max) back and 1/sum to invSum[row]; the
// normalization is applied in the AV GEMM store. Masked (-1e8) key positions
// of the reference underflow to exactly 0 in fp32, so softmax over the HW
// real keys is numerically identical to the reference's 4HW softmax.
// ---------------------------------------------------------------------------
__global__ void __launch_bounds__(256)
softmax_exp_kernel(float* __restrict__ S, float* __restrict__ invSum, int P) {
  float* row = S + (size_t)blockIdx.x * P;
  const int tid = threadIdx.x, lane = tid & 31, wv = tid >> 5;
  __shared__ float sred[8];

  float vb[4];
#pragma unroll
  for (int t = 0; t < 4; ++t) {
    const int i = tid + t * 256;
    vb[t] = (i < P) ? row[i] : -3.402823466e38f;
  }

  float mx = fmaxf(fmaxf(vb[0], vb[1]), fmaxf(vb[2], vb[3]));
  for (int o = 16; o > 0; o >>= 1) mx = fmaxf(mx, __shfl_xor(mx, o, 32));
  if (lane == 0) sred[wv] = mx;
  __syncthreads();
  if (tid < 8) {
    float v = sred[tid];
    for (int o = 4; o > 0; o >>= 1) v = fmaxf(v, __shfl_xor(v, o, 32));
    if (tid == 0) sred[0] = v;
  }
  __syncthreads();
  mx = sred[0];
  __syncthreads();

  float s = 0.f;
#pragma unroll
  for (int t = 0; t < 4; ++t) {
    const int i = tid + t * 256;
    const float e = expf(vb[t] - mx);
    vb[t] = e;
    if (i < P) s += e;
  }
  for (int o = 16; o > 0; o >>= 1) s += __shfl_xor(s, o, 32);
  if (lane == 0) sred[wv] = s;
  __syncthreads();
  if (tid < 8) {
    float v = sred[tid];
    for (int o = 4; o > 0; o >>= 1) v += __shfl_xor(v, o, 32);
    if (tid == 0) sred[0] = v;
  }
  __syncthreads();
  if (tid == 0) invSum[blockIdx.x] = 1.f / sred[0];

#pragma unroll
  for (int t = 0; t < 4; ++t) {
    const int i = tid + t * 256;
    if (i < P) row[i] = vb[t];
  }
}

// ---------------------------------------------------------------------------
// Elementwise / reduction helpers
// ---------------------------------------------------------------------------
__global__ void sigmoid64_kernel(const float* __restrict__ in, float* __restrict__ out, int n) {
  int i = blockIdx.x * blockDim.x + threadIdx.x;
  if (i < n) out[i] = 1.f / (1.f + expf(-64.f * (in[i] - 0.5f)));
}

__global__ void zero_u32_kernel(unsigned int* p) { *p = 0u; }

__global__ void count_ones_kernel(const float* __restrict__ y, int n, unsigned int* cnt) {
  unsigned int c = 0;
  for (int i = blockIdx.x * blockDim.x + threadIdx.x; i < n; i += gridDim.x * blockDim.x)
    c += (y[i] >= 0.5f) ? 1u : 0u;
  for (int o = 16; o > 0; o >>= 1) c += __shfl_xor(c, o, 32);
  __shared__ unsigned int sc[8];
  const int wv = threadIdx.x >> 5, lane = threadIdx.x & 31;
  if (lane == 0) sc[wv] = c;
  __syncthreads();
  if (threadIdx.x == 0) {
    unsigned int t = 0;
    for (int i = 0; i < (int)(blockDim.x >> 5); ++i) t += sc[i];
    atomicAdd(cnt, t);  // integer atomic: deterministic
  }
}

__global__ void entropy_final_kernel(const unsigned int* cnt, float n, float* out) {
  if (threadIdx.x == 0 && blockIdx.x == 0) {
    float p1 = (float)(*cnt) / n;
    float p0 = 1.f - p1;
    const float inv_log2 = 1.4426950408889634f;
    out[0] = (-p1 * logf(p1 + 1e-10f) - p0 * logf(p0 + 1e-10f)) * inv_log2;
  }
}

// ---------------------------------------------------------------------------
// Host orchestration
// ---------------------------------------------------------------------------
extern "C" void kernel_launch(void* const* d_in, const int* in_sizes, int n_in,
                              void* d_out, int out_size, void* d_ws, size_t ws_size,
                              hipStream_t stream) {
  (void)in_sizes; (void)n_in; (void)out_size; (void)ws_size;

  const float* x = (const float*)d_in[0];
  // d_in[1] = idx (unused)
  const float* w_s1 = (const float*)d_in[2];
  const float* e1c1 = (const float*)d_in[3];
  const float* e1c2 = (const float*)d_in[4];
  const float* w_s2 = (const float*)d_in[5];
  const float* e2c1 = (const float*)d_in[6];
  const float* e2c2 = (const float*)d_in[7];
  const float* e3c1 = (const float*)d_in[8];
  const float* e3c2 = (const float*)d_in[9];
  const float* w_s3 = (const float*)d_in[10];
  const float* w_d1 = (const float*)d_in[11];
  const float* w_d2 = (const float*)d_in[12];
  const float* dc1  = (const float*)d_in[13];
  const float* dc2  = (const float*)d_in[14];
  const float* u1q  = (const float*)d_in[15];
  const float* u1k  = (const float*)d_in[16];
  const float* u1v  = (const float*)d_in[17];
  const float* d1c1 = (const float*)d_in[18];
  const float* d1c2 = (const float*)d_in[19];
  const float* u2q  = (const float*)d_in[20];
  const float* u2k  = (const float*)d_in[21];
  const float* u2v  = (const float*)d_in[22];
  const float* d2c1 = (const float*)d_in[23];
  const float* d2c2 = (const float*)d_in[24];
  const float* u3q  = (const float*)d_in[25];
  const float* u3k  = (const float*)d_in[26];
  const float* u3v  = (const float*)d_in[27];
  const float* d3c1 = (const float*)d_in[28];
  const float* d3c2 = (const float*)d_in[29];
  const float* w_d4 = (const float*)d_in[30];

  float* outF = (float*)d_out;

  // -------- workspace bump allocator (floats; all chunks multiple of 4) ----
  float* wsf = (float*)d_ws;
  size_t off = 0;
  auto alloc = [&](size_t n) { float* p = wsf + off; off += n; return p; };

  float* t1  = alloc(4 * 64 * 16 * 16);
  float* h16 = alloc(4 * 64 * 16 * 16);
  float* t2  = alloc(4 * 64 * 16 * 16);
  float* t3  = alloc(4 * 64 * 8 * 8);
  float* h8  = alloc(4 * 64 * 8 * 8);
  float* t4  = alloc(4 * 64 * 8 * 8);
  float* t5  = alloc(4 * 64 * 8 * 8);
  float* y0  = alloc(4 * 32 * 8 * 8);
  float* yb  = alloc(4 * 32 * 8 * 8);
  float* db1 = alloc(4 * 64 * 8 * 8);
  float* db2 = alloc(4 * 64 * 8 * 8);
  float* db3 = alloc(4 * 64 * 8 * 8);
  float* cntf = alloc(4);                  // 16B slot keeps later buffers aligned
  unsigned int* cnt = (unsigned int*)cntf;
  float* qs   = alloc(4 * 8 * 32 * 32);    // [B,8,H,W]  (max stage H=32)
  float* ks   = alloc(4 * 8 * 32 * 32);    // [B,8,P]
  float* ksT  = alloc(4 * 32 * 32 * 8);    // [B,P,8]
  float* vs   = alloc(4 * 64 * 32 * 32);   // [B,64,P]
  float* qupT = alloc(4 * 64 * 64 * 8);    // [B,4HW,8]
  float* invS = alloc(4096);               // per-row 1/sum for one batch
  float* S    = alloc((size_t)4096 * 1024);// one batch of u3 scores, reused
  float* a1  = alloc(4 * 64 * 16 * 16);
  float* d4  = alloc(4 * 64 * 16 * 16);
  float* a2  = alloc(4 * 64 * 32 * 32);
  float* h32 = alloc(4 * 64 * 32 * 32);
  float* d5  = alloc(4 * 64 * 32 * 32);
  float* a3  = alloc(4 * 64 * 64 * 64);
  float* h64 = alloc(4 * 64 * 64 * 64);
  float* d6  = alloc(4 * 64 * 64 * 64);

  // -------- launch helpers --------
  auto conv = [&](const float* in, const float* wgt, const float* resid, float* out,
                  int B, int Cin, int Hin, int Win, int Cout,
                  int KH, int KW, int s, int p, int relu) {
    const int Hout = (Hin + 2 * p - KH) / s + 1;
    const int Wout = (Win + 2 * p - KW) / s + 1;
    const int N = B * Hout * Wout;
    const int tiles = ((Cout + 15) / 16) * ((N + 63) / 64);
    conv_wmma_kernel<<<(tiles + 3) / 4, 128, 0, stream>>>(
        in, wgt, resid, out, B, Cin, Hin, Win, Cout, Hout, Wout, KH, KW, s, p, relu);
  };

  auto gemm_kk = [&](const float* A, const float* Bm, float* C, int M, int N, int K,
                     int lam, int lbn, int lcm, int lcn,
                     const float* colScale, int relu) {
    const int tiles = ((M + 15) / 16) * ((N + 63) / 64);
    gemm_kk_wmma_kernel<<<(tiles + 3) / 4, 128, 0, stream>>>(
        A, Bm, C, M, N, K, lam, lbn, lcm, lcn, colScale, relu);
  };

  auto resblock = [&](const float* xin, const float* w1, const float* w2,
                      float* htmp, float* outp, int H, int W) {
    conv(xin,  w1, nullptr, htmp, 4, 64, H, W, 64, 3, 3, 1, 1, 1);
    conv(htmp, w2, xin,     outp, 4, 64, H, W, 64, 3, 3, 1, 1, 1);
  };

  // attention_conv with zero-upsample/mask sparsity folded in:
  // queries Mq = 4*H*W (bilinear-up q), keys/values P = H*W (even positions).
  auto attention = [&](const float* inp, const float* wq, const float* wk, const float* wv,
                       float* aout, int H, int relu) {
    const int W = H, P = H * W, Mq = 4 * P;
    conv(inp, wq, nullptr, qs, 4, 64, H, W, 8,  1, 1, 1, 0, 0);
    conv(inp, wk, nullptr, ks, 4, 64, H, W, 8,  1, 1, 1, 0, 0);
    conv(inp, wv, nullptr, vs, 4, 64, H, W, 64, 1, 1, 1, 0, 0);
    transpose_cp_kernel<<<(4 * 8 * P + 255) / 256, 256, 0, stream>>>(ks, ksT, 4, 8, P);
    bilinear_up2_t_kernel<<<(4 * 8 * Mq + 255) / 256, 256, 0, stream>>>(qs, qupT, 4, 8, H, W);
    for (int b = 0; b < 4; ++b) {
      const float* qb = qupT + (size_t)b * Mq * 8;   // [Mq][8]
      const float* kb = ksT  + (size_t)b * P * 8;    // [P][8]
      const float* vb = vs   + (size_t)b * 64 * P;   // [64][P]
      float*       ab = aout + (size_t)b * 64 * Mq;  // [64][Mq]
      // scores: S[m,p] = sum_c q[m,c] * k[p,c]   (M=Mq, N=P, K=8)
      gemm_kk(qb, kb, S, Mq, P, 8, 8, 8, P, 1, nullptr, 0);
      // exp(s - rowmax) in place + 1/rowsum
      softmax_exp_kernel<<<Mq, 256, 0, stream>>>(S, invS, P);
      // out[c,m] = invS[m] * sum_p v[c,p] * E[m,p]   (M=64, N=Mq, K=P)
      gemm_kk(vb, S, ab, 64, Mq, P, P, P, Mq, 1, invS, relu);
    }
  };

  // -------- network --------
  conv(x, w_s1, nullptr, t1, 4, 3, 64, 64, 64, 8, 8, 4, 2, 1);      // [4,64,16,16]
  resblock(t1, e1c1, e1c2, h16, t2, 16, 16);
  conv(t2, w_s2, nullptr, t3, 4, 64, 16, 16, 64, 4, 4, 2, 1, 1);    // [4,64,8,8]
  resblock(t3, e2c1, e2c2, h8, t4, 8, 8);
  resblock(t4, e3c1, e3c2, h8, t5, 8, 8);
  conv(t5, w_s3, nullptr, y0, 4, 64, 8, 8, 32, 3, 3, 1, 1, 0);      // [4,32,8,8]

  sigmoid64_kernel<<<(8192 + 255) / 256, 256, 0, stream>>>(y0, yb, 8192);
  zero_u32_kernel<<<1, 1, 0, stream>>>(cnt);
  count_ones_kernel<<<32, 256, 0, stream>>>(yb, 8192, cnt);
  entropy_final_kernel<<<1, 1, 0, stream>>>(cnt, 8192.f, outF + 4 * 3 * 64 * 64);

  conv(yb,  w_d1, nullptr, db1, 4, 32, 8, 8, 64, 3, 3, 1, 1, 1);
  conv(db1, w_d2, nullptr, db2, 4, 64, 8, 8, 64, 3, 3, 1, 1, 1);
  resblock(db2, dc1, dc2, h8, db3, 8, 8);

  attention(db3, u1q, u1k, u1v, a1, 8, 1);       // -> [4,64,16,16], relu
  resblock(a1, d1c1, d1c2, h16, d4, 16, 16);
  attention(d4, u2q, u2k, u2v, a2, 16, 1);       // -> [4,64,32,32], relu
  resblock(a2, d2c1, d2c2, h32, d5, 32, 32);
  attention(d5, u3q, u3k, u3v, a3, 32, 0);       // out_init -> [4,64,64,64]
  resblock(a3, d3c1, d3c2, h64, d6, 64, 64);
  conv(d6, w_d4, nullptr, outF, 4, 64, 64, 64, 3, 3, 3, 1, 1, 0);   // [4,3,64,64] -> d_out
}